// GraphConvLayer_63943473103526
// MI455X (gfx1250) — compile-verified
//
#include <hip/hip_runtime.h>
#include <math.h>

// ---------------------------------------------------------------------------
// GraphConv layer for MI455X (gfx1250, wave32, WMMA).
//   out = silu(concat(feat, mean_k(feat[knn]), mean_k(rel), std_k(rel)) @ W + b)
// N=100000, C=64, K=16, H=128, KDIM=2C+6=134 (padded to 136 for 34 x K=4 WMMA)
// Gather (410MB from L2) is the bound; GEMM (3.4 GFLOP) runs on f32 WMMA.
// ---------------------------------------------------------------------------

typedef float v2f __attribute__((ext_vector_type(2)));
typedef float v4f __attribute__((ext_vector_type(4)));
typedef float v8f __attribute__((ext_vector_type(8)));
typedef int   v4i __attribute__((ext_vector_type(4)));

#define NPTS   100000
#define CCH    64
#define KNB    16
#define HOUT   128
#define KDIM   134      // 2*C + 6
#define KPAD   136      // padded K, multiple of 4
#define KP2    (KPAD/2) // 68 k-pairs
#define MBLK   64       // points per block
#define MIX_LD 140      // dword stride; 140 % 64 == 12 -> A b64 loads tile all 64 banks
                        // row stride = 560B (16B aligned -> b128 stores legal)

// Async DMA: global -> LDS, 16 bytes per lane, tracked by ASYNCcnt (no VGPR
// round trip, no loadcnt stall). lds_off is the wave-relative LDS byte offset
// (= low 32 bits of the flat shared-memory address per the LDS aperture map).
__device__ __forceinline__ void async_copy_b128(unsigned lds_off, const void* gaddr) {
  asm volatile("global_load_async_to_lds_b128 %0, %1, off"
               :: "v"(lds_off), "v"(gaddr) : "memory");
}
__device__ __forceinline__ void wait_asynccnt0() {
#if __has_builtin(__builtin_amdgcn_s_wait_asynccnt)
  __builtin_amdgcn_s_wait_asynccnt(0);
#else
  asm volatile("s_wait_asynccnt 0x0" ::: "memory");
#endif
}

// --- Prologue: pad W (134x128) to 136 rows, swizzle into (k-pair, n) float2 ---
// wp[kp*128 + n] = { W[2kp][n], W[2kp+1][n] }  (zeros past row 133)
// WMMA B operand becomes one b64 load per lane:
//   lanes 0-15  : VGPR0 = W[k0][n0+l],   VGPR1 = W[k0+1][n0+l]
//   lanes 16-31 : VGPR0 = W[k0+2][n0+l], VGPR1 = W[k0+3][n0+l]
__global__ void pad_swizzle_W(const float* __restrict__ W, v2f* __restrict__ wp) {
  int i = blockIdx.x * 256 + threadIdx.x;
  if (i >= KP2 * HOUT) return;
  int kp = i >> 7, n = i & 127;
  int k = 2 * kp;
  float x = (k     < KDIM) ? W[k * HOUT + n]       : 0.0f;
  float y = (k + 1 < KDIM) ? W[(k + 1) * HOUT + n] : 0.0f;
  v2f v; v.x = x; v.y = y;
  wp[i] = v;
}

__global__ __launch_bounds__(256) void graphconv_wmma(
    const float* __restrict__ feat,    // (N, 64)
    const float* __restrict__ coords,  // (N, 3)
    const int*   __restrict__ knn,     // (N, 16)
    const v2f*   __restrict__ wp,      // (68, 128) k-pair swizzled W
    const float* __restrict__ bias,    // (128,)
    float*       __restrict__ out)     // (N, 128)
{
  __shared__ float mix[MBLK][MIX_LD];  // 64 x 136 used (A matrix), bank-tuned stride
  __shared__ int   nidx[MBLK][KNB];

  const int t    = threadIdx.x;
  const int base = blockIdx.x * MBLK;

  // Warm the swizzled-W block (69.6KB) into all cache levels (WGP-scope
  // prefetch) before the MAC loop: it is re-read by every wave every k-step.
  __builtin_prefetch(((const char*)wp) + t * 272, 0, 3);

  // ---- Phase 1a (async): center features -> mix[p][0..63] via ASYNC-to-LDS ----
  // 1024 x b128, issued before the index loads; settled just before phase 2.
  #pragma unroll
  for (int j = 0; j < 4; ++j) {
    int i = t + j * 256;                        // 0..1023 float4 slots
    int p = i >> 4, c4 = (i & 15) * 4;
    int gp = base + p; if (gp >= NPTS) gp = NPTS - 1;
    unsigned lds_off = (unsigned)(uintptr_t)&mix[p][c4];
    async_copy_b128(lds_off, &feat[gp * CCH + c4]);
  }

  // ---- neighbor indices (256 x int4) ----
  {
    int p = t >> 2, k4 = (t & 3) * 4;           // one int4 per thread
    int gp = base + p; if (gp >= NPTS) gp = NPTS - 1;
    *(v4i*)&nidx[p][k4] = *(const v4i*)&knn[gp * KNB + k4];
  }
  __syncthreads();

  // ---- Phase 1b: neighbor feature mean (the dominant L2 gather) ----
  // 16 threads/point x float4: one b128 per neighbor row per lane group.
  {
    const int c4 = (t & 15) * 4;
    const int pg = t >> 4;                      // 16 points per pass
    #pragma unroll
    for (int pass = 0; pass < MBLK / 16; ++pass) {
      const int p = pg + pass * 16;
      v4f s = {0.0f, 0.0f, 0.0f, 0.0f};
      #pragma unroll
      for (int k = 0; k < KNB; ++k) {
        v4f nf = *(const v4f*)&feat[nidx[p][k] * CCH + c4];  // 16 lanes -> 256B row
        s += nf;
      }
      *(v4f*)&mix[p][CCH + c4] = s * (1.0f / KNB);
    }
  }

  // ---- Phase 1c: relative-coordinate mean / population std ----
  if (t < MBLK * 3) {
    const int p = t / 3, d = t - 3 * p;
    int gp = base + p; if (gp >= NPTS) gp = NPTS - 1;
    const float ctr = coords[gp * 3 + d];
    float s = 0.0f, ss = 0.0f;
    #pragma unroll
    for (int k = 0; k < KNB; ++k) {
      float v = coords[nidx[p][k] * 3 + d] - ctr;
      s += v; ss += v * v;
    }
    float m   = s * (1.0f / KNB);
    float var = ss * (1.0f / KNB) - m * m;
    mix[p][2 * CCH + d]     = m;
    mix[p][2 * CCH + 3 + d] = sqrtf(fmaxf(var, 0.0f));
  }
  if (t < MBLK) { mix[t][134] = 0.0f; mix[t][135] = 0.0f; }  // K pad

  wait_asynccnt0();      // settle the async center-feature DMA (this wave)
  __syncthreads();       // ... and make it visible across all waves

  // ---- Phase 2: (64 x 136) @ (136 x 128) via V_WMMA_F32_16X16X4_F32 ----
  // Each wave owns a 16-column strip across ALL 4 row tiles:
  //   B loaded once per k-step (global b64, cache-resident),
  //   A loaded 4x per k-step from LDS (ds_load_b64, conflict-free stride).
  const int w     = t >> 5;                     // wave id: column strip
  const int lane  = t & 31;
  const int lhalf = lane >> 4;                  // 0: lanes 0-15, 1: lanes 16-31
  const int lmod  = lane & 15;
  const int n0    = w * 16;

  v8f acc[4] = {{0}, {0}, {0}, {0}};
  #pragma unroll 2
  for (int kp = 0; kp < KP2 / 2; ++kp) {        // 34 steps of K=4, k0 = 4*kp
    v2f b = wp[(2 * kp + lhalf) * HOUT + n0 + lmod];
    #pragma unroll
    for (int rt = 0; rt < 4; ++rt) {
      v2f a = *(const v2f*)&mix[rt * 16 + lmod][4 * kp + 2 * lhalf];
      acc[rt] = __builtin_amdgcn_wmma_f32_16x16x4_f32(
          false, a, false, b, (short)0, acc[rt], false, false);
    }
  }

  // ---- Epilogue: bias + SiLU, store with C/D layout (VGPR r: M=r / M=r+8) ----
  const float bv = bias[n0 + lmod];
  #pragma unroll
  for (int rt = 0; rt < 4; ++rt) {
    #pragma unroll
    for (int r = 0; r < 8; ++r) {
      const int m  = rt * 16 + r + 8 * lhalf;
      const int gp = base + m;
      if (gp < NPTS) {
        float v = acc[rt][r] + bv;
        float sig = 1.0f / (1.0f + __expf(-v));
        out[gp * HOUT + n0 + lmod] = v * sig;
      }
    }
  }
}

extern "C" void kernel_launch(void* const* d_in, const int* in_sizes, int n_in,
                              void* d_out, int out_size, void* d_ws, size_t ws_size,
                              hipStream_t stream) {
  (void)in_sizes; (void)n_in; (void)out_size; (void)ws_size;
  const float* feat   = (const float*)d_in[0];
  const float* coords = (const float*)d_in[1];
  const int*   knn    = (const int*)d_in[2];
  const float* W      = (const float*)d_in[3];
  const float* b      = (const float*)d_in[4];
  float*       out    = (float*)d_out;
  v2f*         wp     = (v2f*)d_ws;   // needs 68*128*8 = 69,632 bytes of scratch

  pad_swizzle_W<<<(KP2 * HOUT + 255) / 256, 256, 0, stream>>>(W, wp);

  const int nblocks = (NPTS + MBLK - 1) / MBLK;   // 1563
  graphconv_wmma<<<nblocks, 256, 0, stream>>>(feat, coords, knn, wp, b, out);
}